// NGCF_85813446574107
// MI455X (gfx1250) — compile-verified
//
#include <hip/hip_runtime.h>

#define NUM_USERS 100000
#define NUM_ITEMS 200000
#define N_NODES   (NUM_USERS + NUM_ITEMS)   // 300000
#define NNZ       4000000
#define D         64
#define N_LAYERS  3

typedef __attribute__((ext_vector_type(2))) float v2f;
typedef __attribute__((ext_vector_type(8))) float v8f;

// ---------------------------------------------------------------------------
// ego = concat(user_emb, item_emb);  acc (= d_out) = ego  (layer-0 term of mean)
// ---------------------------------------------------------------------------
__global__ void ngcf_init_kernel(const float* __restrict__ user_emb,
                                 const float* __restrict__ item_emb,
                                 float* __restrict__ ego,
                                 float* __restrict__ acc) {
    long long i = (long long)blockIdx.x * blockDim.x + threadIdx.x;
    const long long total = (long long)N_NODES * D;
    if (i >= total) return;
    const long long usplit = (long long)NUM_USERS * D;
    float v = (i < usplit) ? user_emb[i] : item_emb[i - usplit];
    ego[i] = v;
    acc[i] = v;
}

// ---------------------------------------------------------------------------
// SpMM scatter: side[rows[e]] += vals[e] * ego[cols[e]]
// One thread per (edge, float4-chunk). Gathers/scatters hit the 192MB L2
// (ego/side are 76.8MB each), so fp32 atomics to L2 absorb the scatter.
// ---------------------------------------------------------------------------
__global__ void ngcf_spmm_kernel(const float* __restrict__ vals,
                                 const int*   __restrict__ rows,
                                 const int*   __restrict__ cols,
                                 const float* __restrict__ ego,
                                 float*       __restrict__ side) {
    long long i = (long long)blockIdx.x * blockDim.x + threadIdx.x;
    if (i >= (long long)NNZ * 16) return;
    int e = (int)(i >> 4);
    int q = (int)(i & 15);              // which float4 chunk of the 64-dim row
    float v = vals[e];
    int c = cols[e];
    int r = rows[e];
    float4 x = *reinterpret_cast<const float4*>(ego + (long long)c * D + q * 4);
    float* dst = side + (long long)r * D + q * 4;
    atomicAdd(dst + 0, v * x.x);
    atomicAdd(dst + 1, v * x.y);
    atomicAdd(dst + 2, v * x.z);
    atomicAdd(dst + 3, v * x.w);
}

// ---------------------------------------------------------------------------
// Fused per-layer kernel (one wave == one 16-row tile):
//   sum = side @ W1^T ; bi = (ego*side) @ W2^T
//   new_ego = leaky_relu(sum + bi + b1 + b2, 0.2)
//   acc += l2_normalize(new_ego) ; ego = new_ego
// Uses V_WMMA_F32_16X16X4_F32; 16 K-steps x 2 GEMMs x 4 N-tiles = 128 WMMA/wave.
// ---------------------------------------------------------------------------
__global__ __launch_bounds__(256) void ngcf_layer_kernel(
    const float* __restrict__ side,
    float*       __restrict__ ego,
    const float* __restrict__ W1,   // [64,64], row n holds weights for out-col n
    const float* __restrict__ b1,   // [64]
    const float* __restrict__ W2,
    const float* __restrict__ b2,
    float*       __restrict__ acc) {

    const int lane = threadIdx.x & 31;
    const int wave = threadIdx.x >> 5;
    const int tile = blockIdx.x * 8 + wave;          // 8 waves / 256-thread block
    if (tile >= N_NODES / 16) return;                // uniform per wave: EXEC all-1 below
    const int row0 = tile * 16;
    const int m    = lane & 15;                      // A: row-in-tile ; B: col-in-tile
    const int half = lane >> 4;                      // K sub-offset selector

    // ---- preload A fragments (16x4 f32 layout: lane<16 -> K={kb,kb+1}, lane>=16 -> K={kb+2,kb+3})
    v2f aS[16];   // side rows
    v2f aP[16];   // (ego .* side) rows
    const float* srow = side + (long long)(row0 + m) * D;
    const float* erow = ego  + (long long)(row0 + m) * D;
#pragma unroll
    for (int kk = 0; kk < 16; ++kk) {
        const int kb = kk * 4 + half * 2;
        v2f s; s.x = srow[kb]; s.y = srow[kb + 1];
        v2f e; e.x = erow[kb]; e.y = erow[kb + 1];
        aS[kk] = s;
        v2f p; p.x = s.x * e.x; p.y = s.y * e.y;
        aP[kk] = p;
    }

    // ---- per 16-column output tile: two WMMA GEMM chains, then activation
    v8f comb[4];
#pragma unroll
    for (int t = 0; t < 4; ++t) {
        const int col   = t * 16 + m;                // output column this lane owns
        const float* w1r = W1 + (long long)col * D;  // B[k][n] = W[n][k]
        const float* w2r = W2 + (long long)col * D;
        v8f c1 = {};
        v8f c2 = {};
#pragma unroll
        for (int kk = 0; kk < 16; ++kk) {
            const int kb = kk * 4 + half * 2;
            v2f bW1; bW1.x = w1r[kb]; bW1.y = w1r[kb + 1];
            v2f bW2; bW2.x = w2r[kb]; bW2.y = w2r[kb + 1];
            c1 = __builtin_amdgcn_wmma_f32_16x16x4_f32(
                    false, aS[kk], false, bW1, (short)0, c1, false, false);
            c2 = __builtin_amdgcn_wmma_f32_16x16x4_f32(
                    false, aP[kk], false, bW2, (short)0, c2, false, false);
        }
        const float bias = b1[col] + b2[col];
        v8f cc;
#pragma unroll
        for (int r = 0; r < 8; ++r) {
            float s = c1[r] + c2[r] + bias;
            cc[r] = (s >= 0.0f) ? s : 0.2f * s;      // leaky_relu(0.2)
        }
        comb[t] = cc;
    }

    // ---- row-wise L2 norm (D frag: vgpr r holds row r (lanes 0-15) / r+8 (lanes 16-31))
#pragma unroll
    for (int r = 0; r < 8; ++r) {
        float p = 0.0f;
#pragma unroll
        for (int t = 0; t < 4; ++t) { float v = comb[t][r]; p += v * v; }
        // butterfly within each 16-lane half (masks 1,2,4,8 never cross halves)
#pragma unroll
        for (int mask = 1; mask < 16; mask <<= 1) p += __shfl_xor(p, mask, 32);
        float inv = 1.0f / fmaxf(sqrtf(p), 1e-12f);
        const int row = row0 + r + 8 * half;
#pragma unroll
        for (int t = 0; t < 4; ++t) {
            const int col = t * 16 + m;
            const float v = comb[t][r];
            ego[(long long)row * D + col] = v;        // next layer's input (unnormalized)
            acc[(long long)row * D + col] += v * inv; // mean accumulator gets normalized
        }
    }
}

// ---------------------------------------------------------------------------
// out = acc / 4  (mean over {ego0, n(ego1), n(ego2), n(ego3)})
// ---------------------------------------------------------------------------
__global__ void ngcf_finalize_kernel(float* __restrict__ acc) {
    long long i = (long long)blockIdx.x * blockDim.x + threadIdx.x;
    if (i >= (long long)N_NODES * D) return;
    acc[i] *= 0.25f;
}

// ---------------------------------------------------------------------------
extern "C" void kernel_launch(void* const* d_in, const int* in_sizes, int n_in,
                              void* d_out, int out_size, void* d_ws, size_t ws_size,
                              hipStream_t stream) {
    const float* user_emb = (const float*)d_in[0];   // [100000,64]
    const float* item_emb = (const float*)d_in[1];   // [200000,64]
    const float* W1       = (const float*)d_in[2];   // [3,64,64]
    const float* b1       = (const float*)d_in[3];   // [3,64]
    const float* W2       = (const float*)d_in[4];   // [3,64,64]
    const float* b2       = (const float*)d_in[5];   // [3,64]
    const float* vals     = (const float*)d_in[6];   // [4000000]
    const int*   rows     = (const int*)d_in[7];
    const int*   cols     = (const int*)d_in[8];
    float* out = (float*)d_out;                      // [300000,64] == mean accumulator

    const long long nd = (long long)N_NODES * D;     // 19.2M floats
    float* ego  = (float*)d_ws;                      // 76.8 MB
    float* side = ego + nd;                          // 76.8 MB
    (void)in_sizes; (void)n_in; (void)out_size; (void)ws_size;

    const int T = 256;
    const int gInit  = (int)((nd + T - 1) / T);
    const int gSpmm  = (int)(((long long)NNZ * 16 + T - 1) / T);
    const int gLayer = (N_NODES / 16 + 7) / 8;       // 8 waves (tiles) per block

    ngcf_init_kernel<<<gInit, T, 0, stream>>>(user_emb, item_emb, ego, out);

    for (int k = 0; k < N_LAYERS; ++k) {
        hipMemsetAsync(side, 0, nd * sizeof(float), stream);
        ngcf_spmm_kernel<<<gSpmm, T, 0, stream>>>(vals, rows, cols, ego, side);
        ngcf_layer_kernel<<<gLayer, T, 0, stream>>>(
            side, ego,
            W1 + (long long)k * D * D, b1 + (long long)k * D,
            W2 + (long long)k * D * D, b2 + (long long)k * D,
            out);
    }

    ngcf_finalize_kernel<<<gInit, T, 0, stream>>>(out);
}